// DualRoadGNN_13812614824390
// MI455X (gfx1250) — compile-verified
//
#include <hip/hip_runtime.h>

#define N_NODES 65536
#define BGRAPHS 128
#define NPG     512
#define HDIM    128
#define NEDGES  1048576
#define KNN     8
#define EPSN    1e-5f
#define SLOPE   0.01f

typedef float v2f __attribute__((ext_vector_type(2)));
typedef float v8f __attribute__((ext_vector_type(8)));

__device__ __forceinline__ void atomic_add_f32(float* p, float v) {
  __hip_atomic_fetch_add(p, v, __ATOMIC_RELAXED, __HIP_MEMORY_SCOPE_AGENT);
}

// ---------------------------------------------------------------------------
// OUT[rows x 128] = X[rows x 128] @ W[128 x 128] (+ optional bias)
// One wave -> 16 rows x 128 cols, v_wmma_f32_16x16x4_f32 over K.
// grid = rows/16/8 blocks of 256 threads (8 waves).
// ---------------------------------------------------------------------------
__global__ void __launch_bounds__(256)
gemm128_wmma(const float* __restrict__ X, const float* __restrict__ W,
             const float* __restrict__ bias, float* __restrict__ OUT) {
  const int lane = threadIdx.x & 31;
  const int wave = threadIdx.x >> 5;
  const int row0 = (blockIdx.x * 8 + wave) * 16;
  const int m  = lane & 15;
  const int kb = (lane >> 4) * 2;

  v8f acc[8] = {};
  for (int k0 = 0; k0 < HDIM; k0 += 4) {
    v2f a;
    const float* xp = X + (size_t)(row0 + m) * HDIM + k0 + kb;
    a.x = xp[0]; a.y = xp[1];
#pragma unroll
    for (int j = 0; j < 8; ++j) {
      v2f b;
      const float* wp = W + (size_t)(k0 + kb) * HDIM + j * 16 + m;
      b.x = wp[0]; b.y = wp[HDIM];
      acc[j] = __builtin_amdgcn_wmma_f32_16x16x4_f32(
          false, a, false, b, (short)0, acc[j], false, false);
    }
  }
  const int rOut = row0 + (lane >> 4) * 8;
#pragma unroll
  for (int j = 0; j < 8; ++j) {
    const float bv = bias ? bias[j * 16 + m] : 0.0f;
#pragma unroll
    for (int v = 0; v < 8; ++v)
      OUT[(size_t)(rOut + v) * HDIM + j * 16 + m] = acc[j][v] + bv;
  }
}

// ---------------------------------------------------------------------------
// hn = h / max(||h||, 1e-12) per row. One wave per node (32 lanes x float4).
// ---------------------------------------------------------------------------
__global__ void row_normalize(const float* __restrict__ H, float* __restrict__ HN) {
  const int node = (int)((blockIdx.x * (size_t)blockDim.x + threadIdx.x) >> 5);
  const int lane = threadIdx.x & 31;
  if (node >= N_NODES) return;
  float4 v = ((const float4*)(H + (size_t)node * HDIM))[lane];
  float s = v.x * v.x + v.y * v.y + v.z * v.z + v.w * v.w;
#pragma unroll
  for (int o = 16; o > 0; o >>= 1) s += __shfl_xor(s, o, 32);
  const float inv = 1.0f / fmaxf(sqrtf(s), 1e-12f);
  float4 o4 = make_float4(v.x * inv, v.y * inv, v.z * inv, v.w * inv);
  ((float4*)(HN + (size_t)node * HDIM))[lane] = o4;
}

// ---------------------------------------------------------------------------
// Cosine kNN: block = one 16-query tile of one graph. 8 waves each produce a
// 16x64 sim strip via WMMA -> LDS [16][512]; then top-8 per query row.
// ---------------------------------------------------------------------------
__global__ void __launch_bounds__(256)
knn_topk(const float* __restrict__ HN, int* __restrict__ FSRC) {
  __shared__ float sim[16 * NPG];  // 32 KB
  const int g     = blockIdx.x >> 5;   // 32 tiles per graph
  const int qt    = blockIdx.x & 31;
  const int lane  = threadIdx.x & 31;
  const int wave  = threadIdx.x >> 5;
  const int qbase = qt * 16;
  const int m  = lane & 15;
  const int kb = (lane >> 4) * 2;
  const float* Q = HN + ((size_t)g * NPG + qbase) * HDIM;
  const float* C = HN + (size_t)g * NPG * HDIM;

#pragma unroll
  for (int j = 0; j < 4; ++j) {
    const int c0 = wave * 64 + j * 16;
    v8f acc = {};
    for (int k0 = 0; k0 < HDIM; k0 += 4) {
      v2f a;
      const float* qp = Q + (size_t)m * HDIM + k0 + kb;
      a.x = qp[0]; a.y = qp[1];
      v2f b;  // B[k][n] = C[c0+n][k]
      const float* cp = C + (size_t)(c0 + m) * HDIM + k0 + kb;
      b.x = cp[0]; b.y = cp[1];
      acc = __builtin_amdgcn_wmma_f32_16x16x4_f32(
          false, a, false, b, (short)0, acc, false, false);
    }
    const int rBase = (lane >> 4) * 8;
#pragma unroll
    for (int v = 0; v < 8; ++v) {
      const int r = rBase + v;
      const int col = c0 + m;
      float val = acc[v];
      if (qbase + r == col) val -= 2.0f;  // exclude self-loop
      sim[r * NPG + col] = val;
    }
  }
  __syncthreads();
  if (threadIdx.x < 16) {
    const int r = threadIdx.x;
    const int node = g * NPG + qbase + r;
    float* row = sim + r * NPG;
    for (int k = 0; k < KNN; ++k) {
      float best = -1e30f; int bi = 0;
      for (int c = 0; c < NPG; ++c) {
        const float v = row[c];
        if (v > best) { best = v; bi = c; }
      }
      row[bi] = -2e30f;
      FSRC[(size_t)node * KNN + k] = g * NPG + bi;
    }
  }
}

// ---------------------------------------------------------------------------
// Degrees of spatial graph: deg = 1 + #in-edges; dinv = rsqrt(deg) in-place.
// ---------------------------------------------------------------------------
__global__ void deg_init(float* __restrict__ deg) {
  const int i = blockIdx.x * blockDim.x + threadIdx.x;
  if (i < N_NODES) deg[i] = 1.0f;
}
__global__ void deg_count(const int* __restrict__ ei, float* __restrict__ deg) {
  const int e = blockIdx.x * blockDim.x + threadIdx.x;
  if (e < NEDGES) atomic_add_f32(&deg[ei[NEDGES + e]], 1.0f);
}
__global__ void deg_finalize(float* __restrict__ deg) {
  const int i = blockIdx.x * blockDim.x + threadIdx.x;
  if (i < N_NODES) deg[i] = rsqrtf(deg[i]);
}

// ---------------------------------------------------------------------------
// h1 = dinv^2 * hw + bias   (self-loop + bias init before scatter)
// ---------------------------------------------------------------------------
__global__ void gcn_self_bias(const float* __restrict__ HW, const float* __restrict__ dinv,
                              const float* __restrict__ bias, float* __restrict__ OUT) {
  const size_t idx = blockIdx.x * (size_t)blockDim.x + threadIdx.x;
  const int n = (int)(idx >> 7), c = (int)(idx & 127);
  const float di = dinv[n];
  OUT[idx] = di * di * HW[idx] + bias[c];
}

// ---------------------------------------------------------------------------
// Spatial scatter: wave per edge, lane handles 4 channels (float4 gather,
// 4 agent-scope fadd atomics). This is the memory-bound hot loop (~1 GB/layer).
// ---------------------------------------------------------------------------
__global__ void __launch_bounds__(256)
gcn_scatter(const float* __restrict__ HW, const int* __restrict__ ei,
            const float* __restrict__ dinv, float* __restrict__ OUT) {
  const int e = (int)((blockIdx.x * (size_t)blockDim.x + threadIdx.x) >> 5);
  const int lane = threadIdx.x & 31;
  if (e >= NEDGES) return;
  const int s = ei[e], d = ei[NEDGES + e];
  const float coeff = dinv[s] * dinv[d];
  const float4 v = ((const float4*)(HW + (size_t)s * HDIM))[lane];
  float* dst = OUT + (size_t)d * HDIM + lane * 4;
  atomic_add_f32(dst + 0, coeff * v.x);
  atomic_add_f32(dst + 1, coeff * v.y);
  atomic_add_f32(dst + 2, coeff * v.z);
  atomic_add_f32(dst + 3, coeff * v.w);
}

// ---------------------------------------------------------------------------
// kNN conv aggregation: every node has exactly K=8 in-edges + self => deg=9
// everywhere => pure gather, coefficient 1/9, no atomics.
// ---------------------------------------------------------------------------
__global__ void __launch_bounds__(256)
knn_gather(const float* __restrict__ HW, const int* __restrict__ FSRC,
           const float* __restrict__ bias, float* __restrict__ OUT) {
  const int n = (int)((blockIdx.x * (size_t)blockDim.x + threadIdx.x) >> 5);
  const int lane = threadIdx.x & 31;
  if (n >= N_NODES) return;
  float4 acc = ((const float4*)(HW + (size_t)n * HDIM))[lane];
#pragma unroll
  for (int k = 0; k < KNN; ++k) {
    const int s = FSRC[(size_t)n * KNN + k];
    const float4 v = ((const float4*)(HW + (size_t)s * HDIM))[lane];
    acc.x += v.x; acc.y += v.y; acc.z += v.z; acc.w += v.w;
  }
  const float inv9 = 1.0f / 9.0f;
  const int c = lane * 4;
  float4 o = make_float4(acc.x * inv9 + bias[c + 0], acc.y * inv9 + bias[c + 1],
                         acc.z * inv9 + bias[c + 2], acc.w * inv9 + bias[c + 3]);
  ((float4*)(OUT + (size_t)n * HDIM))[lane] = o;
}

// ---------------------------------------------------------------------------
// GraphNorm stats per (graph, channel): A = mean*ms, R = 1/sqrt(var+eps).
// batch is contiguous blocks of 512 nodes. Block = graph, thread = channel.
// ---------------------------------------------------------------------------
__global__ void __launch_bounds__(128)
norm_stats(const float* __restrict__ V, const float* __restrict__ ms,
           float* __restrict__ A, float* __restrict__ R) {
  const int g = blockIdx.x, c = threadIdx.x;
  const float* p = V + (size_t)g * NPG * HDIM + c;
  float s1 = 0.0f, s2 = 0.0f;
  for (int n = 0; n < NPG; ++n) {
    const float v = p[(size_t)n * HDIM];
    s1 += v; s2 += v * v;
  }
  const float mean = s1 * (1.0f / NPG);
  const float a = mean * ms[c];
  const float var = s2 * (1.0f / NPG) - 2.0f * a * mean + a * a;
  A[g * HDIM + c] = a;
  R[g * HDIM + c] = rsqrtf(var + EPSN);
}

// y = leaky_relu(w*(v-A)*R + b), in place
__global__ void norm_apply_lrelu(float* __restrict__ V, const float* __restrict__ A,
                                 const float* __restrict__ R, const float* __restrict__ w,
                                 const float* __restrict__ b) {
  const size_t idx = blockIdx.x * (size_t)blockDim.x + threadIdx.x;
  const int c = (int)(idx & 127), g = (int)(idx >> 16);  // idx>>7 (node) >>9 (graph)
  float y = w[c] * (V[idx] - A[g * HDIM + c]) * R[g * HDIM + c] + b[c];
  V[idx] = y > 0.0f ? y : SLOPE * y;
}

// h = 0.5*(h1 + leaky_relu(norm(f)))
__global__ void fnorm_combine(const float* __restrict__ F, const float* __restrict__ H1,
                              const float* __restrict__ A, const float* __restrict__ R,
                              const float* __restrict__ w, const float* __restrict__ b,
                              float* __restrict__ Hout) {
  const size_t idx = blockIdx.x * (size_t)blockDim.x + threadIdx.x;
  const int c = (int)(idx & 127), g = (int)(idx >> 16);
  float y = w[c] * (F[idx] - A[g * HDIM + c]) * R[g * HDIM + c] + b[c];
  y = y > 0.0f ? y : SLOPE * y;
  Hout[idx] = 0.5f * (H1[idx] + y);
}

// out[g,c] (+)= scale * mean over the graph's 512 nodes
__global__ void __launch_bounds__(128)
graph_mean(const float* __restrict__ H, float* __restrict__ OUT,
           float scale, int accumulate) {
  const int g = blockIdx.x, c = threadIdx.x;
  const float* p = H + (size_t)g * NPG * HDIM + c;
  float s = 0.0f;
  for (int n = 0; n < NPG; ++n) s += p[(size_t)n * HDIM];
  const float v = scale * s * (1.0f / NPG);
  if (accumulate) OUT[g * HDIM + c] += v;
  else            OUT[g * HDIM + c]  = v;
}

// ---------------------------------------------------------------------------
extern "C" void kernel_launch(void* const* d_in, const int* in_sizes, int n_in,
                              void* d_out, int out_size, void* d_ws, size_t ws_size,
                              hipStream_t stream) {
  const float* x       = (const float*)d_in[0];
  const int*   ei      = (const int*)d_in[1];   // [2, E]
  // d_in[2] = batch (contiguous blocks of 512 -> unused)
  const float* emb_w   = (const float*)d_in[3];
  const float* emb_b   = (const float*)d_in[4];
  const float* conv_w  = (const float*)d_in[5];
  const float* conv_b  = (const float*)d_in[6];
  const float* fconv_w = (const float*)d_in[7];
  const float* fconv_b = (const float*)d_in[8];
  const float* norm_w  = (const float*)d_in[9];
  const float* norm_b  = (const float*)d_in[10];
  const float* norm_ms = (const float*)d_in[11];
  const float* fnorm_w  = (const float*)d_in[12];
  const float* fnorm_b  = (const float*)d_in[13];
  const float* fnorm_ms = (const float*)d_in[14];
  float* out = (float*)d_out;

  const size_t NH = (size_t)N_NODES * HDIM;  // 8M floats
  float* ws   = (float*)d_ws;
  float* h    = ws;            // persistent node features
  float* hw   = ws + NH;       // GEMM output scratch
  float* h1   = ws + 2 * NH;   // also used as hn before layer loop
  float* f    = ws + 3 * NH;
  float* dinv = ws + 4 * NH;                       // N floats
  int*   fsrc = (int*)(ws + 4 * NH + N_NODES);     // N*K ints
  float* A    = (float*)(fsrc + (size_t)N_NODES * KNN);
  float* R    = A + BGRAPHS * HDIM;

  const int gemmBlocks = N_NODES / 16 / 8;  // 512
  const int nhBlocks   = (int)(NH / 256);   // 32768

  // 1. embedding: h = x @ emb_w + emb_b
  gemm128_wmma<<<gemmBlocks, 256, 0, stream>>>(x, emb_w, emb_b, h);

  // 2. cosine kNN graph on h (hn staged in h1)
  row_normalize<<<N_NODES / 8, 256, 0, stream>>>(h, h1);
  knn_topk<<<BGRAPHS * 32, 256, 0, stream>>>(h1, fsrc);

  // 3. spatial degrees -> dinv
  deg_init<<<N_NODES / 256, 256, 0, stream>>>(dinv);
  deg_count<<<NEDGES / 256, 256, 0, stream>>>(ei, dinv);
  deg_finalize<<<N_NODES / 256, 256, 0, stream>>>(dinv);

  // 4. two layers
  for (int i = 0; i < 2; ++i) {
    // spatial GCNConv
    gemm128_wmma<<<gemmBlocks, 256, 0, stream>>>(h, conv_w + (size_t)i * HDIM * HDIM,
                                                 nullptr, hw);
    gcn_self_bias<<<nhBlocks, 256, 0, stream>>>(hw, dinv, conv_b + i * HDIM, h1);
    gcn_scatter<<<NEDGES / 8, 256, 0, stream>>>(hw, ei, dinv, h1);
    norm_stats<<<BGRAPHS, 128, 0, stream>>>(h1, norm_ms + i * HDIM, A, R);
    norm_apply_lrelu<<<nhBlocks, 256, 0, stream>>>(h1, A, R, norm_w + i * HDIM,
                                                   norm_b + i * HDIM);
    // feature (kNN) GCNConv: deg == 9 everywhere -> gather form
    gemm128_wmma<<<gemmBlocks, 256, 0, stream>>>(h1, fconv_w + (size_t)i * HDIM * HDIM,
                                                 nullptr, hw);
    knn_gather<<<N_NODES / 8, 256, 0, stream>>>(hw, fsrc, fconv_b + i * HDIM, f);
    norm_stats<<<BGRAPHS, 128, 0, stream>>>(f, fnorm_ms + i * HDIM, A, R);
    fnorm_combine<<<nhBlocks, 256, 0, stream>>>(f, h1, A, R, fnorm_w + i * HDIM,
                                                fnorm_b + i * HDIM, h);
    // gf = 1*mean(all_h[0]) + 2*mean(all_h[1])   (the all_h[i-1] indexing quirk)
    graph_mean<<<BGRAPHS, 128, 0, stream>>>(h, out, i == 0 ? 1.0f : 2.0f, i);
  }
}